// GNN_25580825215696
// MI455X (gfx1250) — compile-verified
//
#include <hip/hip_runtime.h>

// Hetero-GraphSAGE (2 layers, R=4 etypes, D=10) + edge dot-product scoring.
// MI455X plan: all hot buffers (padded h: 24MB, msum: 24MB, deg: 2MB) are
// L2-resident (192MB), so gather + f32-atomic scatter run at L2 bandwidth.
// Dense 10x10 transforms: f32 WMMA (V_WMMA_F32_16X16X4_F32) on 16-node tiles,
// 3 chained K=4 steps over a zero-padded K=12 layout so every operand load is
// an unconditional global_load_b64 and C/D tails are branch-free clauses.

#define D  10
#define RT 4     // edge types (fixed by the reference)
#define SP 12    // padded feature stride (covers K chunks 0..11, cols 10/11 = 0)

typedef __attribute__((ext_vector_type(2))) float v2f;
typedef __attribute__((ext_vector_type(8))) float v8f;

__device__ __forceinline__ v8f wmma_k4(v2f a, v2f b, v8f c) {
  // D = A(16x4 f32) * B(4x16 f32) + C(16x16 f32)
  return __builtin_amdgcn_wmma_f32_16x16x4_f32(
      /*neg_a=*/false, a, /*neg_b=*/false, b,
      /*c_mod=*/(short)0, c, /*reuse_a=*/false, /*reuse_b=*/false);
}

// ---- prep: zero-padded weight tiles [16][SP]; W_self & bias pre-summed over r.
__global__ void k_prep(const float* __restrict__ Ws1, const float* __restrict__ Wn1,
                       const float* __restrict__ b1,
                       const float* __restrict__ Ws2, const float* __restrict__ Wn2,
                       const float* __restrict__ b2,
                       float* __restrict__ WsP1, float* __restrict__ WnP1,
                       float* __restrict__ bsP1,
                       float* __restrict__ WsP2, float* __restrict__ WnP2,
                       float* __restrict__ bsP2) {
  for (int i = threadIdx.x; i < 16 * SP; i += blockDim.x) {
    int n = i / SP, k = i % SP;
    float s1 = 0.f, s2 = 0.f;
    if (n < D && k < D) {
#pragma unroll
      for (int r = 0; r < RT; ++r) {
        s1 += Ws1[r * D * D + n * D + k];
        s2 += Ws2[r * D * D + n * D + k];
      }
    }
    WsP1[i] = s1;
    WsP2[i] = s2;
  }
  for (int i = threadIdx.x; i < RT * 16 * SP; i += blockDim.x) {
    int r = i / (16 * SP), j = i % (16 * SP);
    int n = j / SP, k = j % SP;
    float v1 = 0.f, v2 = 0.f;
    if (n < D && k < D) {
      v1 = Wn1[r * D * D + n * D + k];
      v2 = Wn2[r * D * D + n * D + k];
    }
    WnP1[i] = v1;
    WnP2[i] = v2;
  }
  for (int i = threadIdx.x; i < 16; i += blockDim.x) {
    float s1 = 0.f, s2 = 0.f;
    if (i < D) {
#pragma unroll
      for (int r = 0; r < RT; ++r) {
        s1 += b1[r * D + i];
        s2 += b2[r * D + i];
      }
    }
    bsP1[i] = s1;
    bsP2[i] = s2;
  }
}

// ---- copy raw [N][D] features into padded [N][SP] layout (cols >= D are 0).
__global__ void k_padcopy(const float* __restrict__ x, float* __restrict__ hp, int Nn) {
  int i = blockIdx.x * blockDim.x + threadIdx.x;
  if (i >= Nn * SP) return;
  int n = i / SP, k = i - n * SP;
  hp[i] = (k < D) ? x[(long long)n * D + k] : 0.0f;
}

__global__ void k_zero(float* __restrict__ p, int n) {
  int i = blockIdx.x * blockDim.x + threadIdx.x;
  if (i < n) p[i] = 0.0f;
}

// One thread per (edge, feature): coalesced 40B gather + 40B atomic scatter.
__global__ void k_scatter(const float* __restrict__ hp,
                          const int* __restrict__ src,
                          const int* __restrict__ dst,
                          float* __restrict__ msum,
                          float* __restrict__ deg, int E_) {
  long long gid = (long long)blockIdx.x * blockDim.x + threadIdx.x;
  if (gid >= (long long)E_ * D) return;
  int e = (int)(gid / D);
  int k = (int)(gid - (long long)e * D);
  int s = src[e];
  int d = dst[e];
  atomicAdd(&msum[(long long)d * SP + k], hp[(long long)s * SP + k]);
  if (k == 0) atomicAdd(&deg[d], 1.0f);
}

// WMMA f32 layouts per ISA 7.12.2 (wave32), one wave per 16-node tile:
//   A: lane -> row M=lane&15, K = c*4 + (lane>>4)*2 + {x,y}
//   B: lane -> col N=lane&15, same K split; B[k][n] = Wpad[n*SP + k]
//   C/D: vgpr j -> row j + (lane>>4)*8, col lane&15
// Accumulator columns n>=10 are identically zero (zero bias pad, zero B cols),
// so stores need no value select and cols 10/11 in memory stay 0.

// h_out = bias_sum + h_in @ (sum_r W_self[r])^T
__global__ void k_self_wmma(const float* __restrict__ hp,   // [N][SP]
                            const float* __restrict__ WsP,  // [16][SP]
                            const float* __restrict__ bsP,  // [16]
                            float* __restrict__ hout,       // [N][SP]
                            int nTiles, int Nn) {
  int wave = blockIdx.x * (blockDim.x >> 5) + (threadIdx.x >> 5);
  if (wave >= nTiles) return;               // wave-uniform
  int lane = threadIdx.x & 31;
  int m    = lane & 15;
  int half = lane >> 4;
  int node = min(wave * 16 + m, Nn - 1);    // clamp: loads stay unconditional

  union { v8f v; float f[8]; } acc;
  float bsum = bsP[m];
#pragma unroll
  for (int j = 0; j < 8; ++j) acc.f[j] = bsum;

  const float* arow = hp + (long long)node * SP;
  const float* brow = WsP + m * SP;
#pragma unroll
  for (int c = 0; c < 3; ++c) {
    int k0 = c * 4 + half * 2;
    v2f a, b;
    a.x = arow[k0]; a.y = arow[k0 + 1];     // global_load_b64, padded zeros
    b.x = brow[k0]; b.y = brow[k0 + 1];
    acc.v = wmma_k4(a, b, acc.v);
  }

  int rowBase = wave * 16 + half * 8;
  bool fullTile = (wave * 16 + 16) <= Nn;   // wave-uniform
  if (m < SP) {
    float* op = hout + (long long)rowBase * SP + m;
    if (fullTile) {
#pragma unroll
      for (int j = 0; j < 8; ++j) op[j * SP] = acc.f[j];   // const-offset clause
    } else {
#pragma unroll
      for (int j = 0; j < 8; ++j)
        if (rowBase + j < Nn) op[j * SP] = acc.f[j];
    }
  }
}

// h_out += (msum / max(deg,1)) @ W_neigh_r^T   (C reloaded from h_out)
__global__ void k_neigh_wmma(const float* __restrict__ msum,  // [N][SP]
                             const float* __restrict__ deg,   // [N]
                             const float* __restrict__ WnP,   // [16][SP] (one r)
                             float* __restrict__ hout,        // [N][SP]
                             int nTiles, int Nn) {
  int wave = blockIdx.x * (blockDim.x >> 5) + (threadIdx.x >> 5);
  if (wave >= nTiles) return;
  int lane = threadIdx.x & 31;
  int m    = lane & 15;
  int half = lane >> 4;
  int node = min(wave * 16 + m, Nn - 1);
  int mc   = min(m, SP - 1);                // lanes 12..15 read col 11 (== 0)
  int rowBase = wave * 16 + half * 8;
  bool fullTile = (wave * 16 + 16) <= Nn;   // wave-uniform

  float rdeg = 1.0f / fmaxf(deg[node], 1.0f);

  union { v8f v; float f[8]; } acc;
  if (fullTile) {
    const float* cp = hout + (long long)rowBase * SP + mc;
#pragma unroll
    for (int j = 0; j < 8; ++j) acc.f[j] = cp[j * SP];     // const-offset clause
  } else {
#pragma unroll
    for (int j = 0; j < 8; ++j)
      acc.f[j] = hout[(long long)min(rowBase + j, Nn - 1) * SP + mc];
  }

  const float* arow = msum + (long long)node * SP;
  const float* brow = WnP + m * SP;
#pragma unroll
  for (int c = 0; c < 3; ++c) {
    int k0 = c * 4 + half * 2;
    v2f a, b;
    a.x = arow[k0] * rdeg; a.y = arow[k0 + 1] * rdeg;
    b.x = brow[k0];        b.y = brow[k0 + 1];
    acc.v = wmma_k4(a, b, acc.v);
  }

  if (m < D) {                              // cols 10/11 already 0 in memory
    float* op = hout + (long long)rowBase * SP + m;
    if (fullTile) {
#pragma unroll
      for (int j = 0; j < 8; ++j) op[j * SP] = acc.f[j];
    } else {
#pragma unroll
      for (int j = 0; j < 8; ++j)
        if (rowBase + j < Nn) op[j * SP] = acc.f[j];
    }
  }
}

// out[0..E) = dot(h[src0],h[dst0]); out[E..2E) = dot(h[nsrc],h[ndst])
__global__ void k_score(const float* __restrict__ hp,  // [N][SP]
                        const int* __restrict__ ps, const int* __restrict__ pd,
                        const int* __restrict__ ns, const int* __restrict__ nd,
                        float* __restrict__ out, int E_) {
  int i = blockIdx.x * blockDim.x + threadIdx.x;
  if (i >= 2 * E_) return;
  int a, b;
  if (i < E_) { a = ps[i]; b = pd[i]; }
  else        { a = ns[i - E_]; b = nd[i - E_]; }
  const float* pa = hp + (long long)a * SP;
  const float* pb = hp + (long long)b * SP;
  float s = 0.0f;
#pragma unroll
  for (int k = 0; k < D; ++k) s += pa[k] * pb[k];
  out[i] = s;
}

extern "C" void kernel_launch(void* const* d_in, const int* in_sizes, int n_in,
                              void* d_out, int out_size, void* d_ws, size_t ws_size,
                              hipStream_t stream) {
  (void)n_in; (void)out_size; (void)ws_size;
  const float* x    = (const float*)d_in[0];
  const int*   esrc = (const int*)d_in[1];   // [RT][E]
  const int*   edst = (const int*)d_in[2];   // [RT][E]
  const int*   nsrc = (const int*)d_in[3];   // [E]
  const int*   ndst = (const int*)d_in[4];   // [E]
  const float* Ws1  = (const float*)d_in[5];
  const float* Wn1  = (const float*)d_in[6];
  const float* b1   = (const float*)d_in[7];
  const float* Ws2  = (const float*)d_in[8];
  const float* Wn2  = (const float*)d_in[9];
  const float* b2   = (const float*)d_in[10];

  const int Nn = in_sizes[0] / D;
  const int Ee = in_sizes[3];

  // Workspace (floats): B0 (h0 / h2) | B1 (h1) | msum | deg | padded weights
  float* B0   = (float*)d_ws;                 // N*SP
  float* B1   = B0 + (size_t)Nn * SP;         // N*SP
  float* msum = B1 + (size_t)Nn * SP;         // N*SP
  float* dega = msum + (size_t)Nn * SP;       // N (contiguous after msum)
  float* WsP1 = dega + (size_t)Nn;            // 16*SP
  float* WnP1 = WsP1 + 16 * SP;               // RT*16*SP
  float* bsP1 = WnP1 + RT * 16 * SP;          // 16
  float* WsP2 = bsP1 + 16;
  float* WnP2 = WsP2 + 16 * SP;
  float* bsP2 = WnP2 + RT * 16 * SP;

  const int tiles   = (Nn + 15) / 16;
  const int wBlocks = (tiles + 3) / 4;        // 4 waves (128 thr) per block
  const int zeroCnt = Nn * SP + Nn;           // msum + deg in one shot
  const int padCnt  = Nn * SP;
  const long long sThreads = (long long)Ee * D;
  const int sBlocks = (int)((sThreads + 255) / 256);

  k_prep<<<1, 256, 0, stream>>>(Ws1, Wn1, b1, Ws2, Wn2, b2,
                                WsP1, WnP1, bsP1, WsP2, WnP2, bsP2);
  k_padcopy<<<(padCnt + 255) / 256, 256, 0, stream>>>(x, B0, Nn);

  auto layer = [&](const float* hin, const float* WsP, const float* WnP,
                   const float* bsP, float* hout) {
    k_self_wmma<<<wBlocks, 128, 0, stream>>>(hin, WsP, bsP, hout, tiles, Nn);
    for (int r = 0; r < RT; ++r) {
      k_zero<<<(zeroCnt + 255) / 256, 256, 0, stream>>>(msum, zeroCnt);
      k_scatter<<<sBlocks, 256, 0, stream>>>(hin, esrc + (size_t)r * Ee,
                                             edst + (size_t)r * Ee, msum, dega, Ee);
      k_neigh_wmma<<<wBlocks, 128, 0, stream>>>(msum, dega, WnP + r * 16 * SP,
                                                hout, tiles, Nn);
    }
  };

  layer(B0, WsP1, WnP1, bsP1, B1);   // layer 1: B0 -> B1
  layer(B1, WsP2, WnP2, bsP2, B0);   // layer 2: B1 -> B0 (reuses h0 buffer)

  k_score<<<(2 * Ee + 255) / 256, 256, 0, stream>>>(B0, esrc, edst, nsrc, ndst,
                                                    (float*)d_out, Ee);
}